// OtsuThreshold_55877524521192
// MI455X (gfx1250) — compile-verified
//
#include <hip/hip_runtime.h>
#include <hip/hip_bf16.h>
#include <math.h>

typedef __attribute__((ext_vector_type(2))) float v2f;
typedef __attribute__((ext_vector_type(8))) float v8f;

#define NROW   48            // B*C = 16*3
#define HW     (1u << 20)    // 1024*1024
#define NTOT   (50331648u)   // 48 * 1M
#define NBINS  256
#define CHUNKS 64            // histogram blocks per row
#define CHUNK_ELEMS 16384    // HW / CHUNKS
#define TILE_FLOATS 4096     // 16KB async-staged LDS tile

// ---------------- float <-> order-preserving uint ----------------
__device__ __forceinline__ unsigned f2ord(float f) {
    unsigned b = __float_as_uint(f);
    return (b & 0x80000000u) ? ~b : (b | 0x80000000u);
}
__device__ __forceinline__ float ord2f(unsigned u) {
    unsigned b = (u & 0x80000000u) ? (u & 0x7fffffffu) : ~u;
    return __uint_as_float(b);
}

// ---------------- 0: init workspace ----------------
__global__ void k_init(unsigned* ws, int n) {
    int i = blockIdx.x * blockDim.x + threadIdx.x;
    int stride = gridDim.x * blockDim.x;
    for (; i < n; i += stride) {
        unsigned v = 0u;
        if (i == 0) v = 0xFFFFFFFFu;   // running min (ordered space)
        ws[i] = v;
    }
}

// ---------------- 1: global min/max ----------------
__global__ void k_minmax(const float4* __restrict__ x4, unsigned* __restrict__ mmx, int n4) {
    __shared__ float smn[256], smx[256];
    int tid = threadIdx.x;
    float lmn =  INFINITY, lmx = -INFINITY;
    int stride = gridDim.x * blockDim.x;
    for (int i = blockIdx.x * blockDim.x + tid; i < n4; i += stride) {
        float4 v = x4[i];
        lmn = fminf(lmn, fminf(fminf(v.x, v.y), fminf(v.z, v.w)));
        lmx = fmaxf(lmx, fmaxf(fmaxf(v.x, v.y), fmaxf(v.z, v.w)));
    }
    smn[tid] = lmn; smx[tid] = lmx;
    __syncthreads();
    for (int s = 128; s > 0; s >>= 1) {
        if (tid < s) {
            smn[tid] = fminf(smn[tid], smn[tid + s]);
            smx[tid] = fmaxf(smx[tid], smx[tid + s]);
        }
        __syncthreads();
    }
    if (tid == 0) {
        atomicMin(&mmx[0], f2ord(smn[0]));
        atomicMax(&mmx[1], f2ord(smx[0]));
    }
}

// ---------------- 2: per-row histogram (async global->LDS staging) ----------------
__global__ void k_hist(const float* __restrict__ x, const unsigned* __restrict__ mmx,
                       unsigned* __restrict__ hist) {
    __shared__ float tile[TILE_FLOATS];          // 16 KB async-staged tile
    __shared__ unsigned sh[8][NBINS];            // per-wave sub-histograms (8 KB)
    const int tid  = threadIdx.x;                // 0..255 (8 waves of 32)
    const int wave = tid >> 5;
    const int row  = blockIdx.y;

    for (int i = tid; i < 8 * NBINS; i += 256) ((unsigned*)sh)[i] = 0u;

    const float mnv = ord2f(mmx[0]);
    const float mxv = ord2f(mmx[1]);
    const float scale = (mxv > mnv) ? ((float)NBINS / (mxv - mnv)) : 0.0f;

    const float* src = x + ((size_t)row << 20) + (size_t)blockIdx.x * CHUNK_ELEMS;
    const unsigned ldsBase = (unsigned)(uintptr_t)&tile[0];  // LDS aperture: low 32 bits = LDS offset
    __syncthreads();

    for (int t = 0; t < CHUNK_ELEMS / TILE_FLOATS; ++t) {
        const float* ts = src + t * TILE_FLOATS;
        // async DMA: 256 threads x 4 x b128 = 16 KB into LDS (tracked by ASYNCcnt)
        #pragma unroll
        for (int k = 0; k < 4; ++k) {
            unsigned loff = ldsBase + (unsigned)((((k << 8) + tid) << 4));
            unsigned long long ga = (unsigned long long)(uintptr_t)(ts + (((k << 8) + tid) << 2));
            asm volatile("global_load_async_to_lds_b128 %0, %1, off"
                         :: "v"(loff), "v"(ga) : "memory");
        }
        asm volatile("s_wait_asynccnt 0x0" ::: "memory");
        __syncthreads();
        // bin out of LDS into this wave's private histogram
        #pragma unroll
        for (int k = 0; k < 4; ++k) {
            float4 v = *reinterpret_cast<const float4*>(&tile[(((k << 8) + tid) << 2)]);
            int b0 = min(max((int)floorf((v.x - mnv) * scale), 0), NBINS - 1);
            int b1 = min(max((int)floorf((v.y - mnv) * scale), 0), NBINS - 1);
            int b2 = min(max((int)floorf((v.z - mnv) * scale), 0), NBINS - 1);
            int b3 = min(max((int)floorf((v.w - mnv) * scale), 0), NBINS - 1);
            atomicAdd(&sh[wave][b0], 1u);
            atomicAdd(&sh[wave][b1], 1u);
            atomicAdd(&sh[wave][b2], 1u);
            atomicAdd(&sh[wave][b3], 1u);
        }
        __syncthreads();
    }
    unsigned sum = 0;
    #pragma unroll
    for (int w = 0; w < 8; ++w) sum += sh[w][tid];
    atomicAdd(&hist[row * NBINS + tid], sum);
}

// ---------------- 3: Otsu via WMMA triangular-matmul cumsum ----------------
// cumsum(h)[n] = sum_{k<=n} h[k]  ==  h(16x256) x U(256x256), U[k][n] = (k<=n)
// One wave per 16-row tile; V_WMMA_F32_16X16X4_F32 accumulates in exact fp32.
__global__ void __launch_bounds__(32) k_otsu(const unsigned* __restrict__ hist,
                                             const unsigned* __restrict__ mmx,
                                             float* __restrict__ thrWs,
                                             float* __restrict__ outThr) {
    __shared__ float sh_h [16][NBINS];
    __shared__ float sh_ht[16][NBINS];
    __shared__ float sh_w [16][NBINS];
    __shared__ float sh_s [16][NBINS];

    const int lane = threadIdx.x;          // 0..31
    const int rt   = blockIdx.x;           // row tile: rows rt*16 .. rt*16+15
    const float inv = 1.0f / (float)HW;

    for (int i = lane; i < 16 * NBINS; i += 32) {
        int m = i >> 8, n = i & 255;
        float h = (float)hist[(rt * 16 + m) * NBINS + n] * inv;
        sh_h [m][n] = h;
        sh_ht[m][n] = h * (float)n;
    }
    __syncthreads();

    // Fragment geometry (ISA 7.12.2, 32-bit 16x4 A / 4x16 B / 16x16 C-D):
    const int mrow  = lane & 15;           // A: M index / B: N index within tile
    const int khalf = (lane >> 4) << 1;    // 0 (lanes 0-15) or 2 (lanes 16-31)
    const int mbase = (lane < 16) ? 0 : 8; // D: VGPR v -> M = v + mbase

    for (int mat = 0; mat < 2; ++mat) {
        const float* H = mat ? &sh_ht[0][0] : &sh_h[0][0];
        float*       O = mat ? &sh_s [0][0] : &sh_w[0][0];
        for (int j = 0; j < 16; ++j) {           // output column tile (bins 16j..16j+15)
            v8f c = {};
            const int kcEnd = 4 * (j + 1);       // upper-triangular: K <= 16j+15 only
            for (int kc = 0; kc < kcEnd; ++kc) {
                const int k0 = 4 * kc + khalf;
                v2f a, b;
                a.x = H[mrow * NBINS + k0];
                a.y = H[mrow * NBINS + k0 + 1];
                const int n = 16 * j + mrow;     // global column for this lane
                b.x = (k0     <= n) ? 1.0f : 0.0f;
                b.y = (k0 + 1 <= n) ? 1.0f : 0.0f;
                c = __builtin_amdgcn_wmma_f32_16x16x4_f32(
                        /*neg_a=*/false, a, /*neg_b=*/false, b,
                        /*c_mod=*/(short)0, c, /*reuse_a=*/false, /*reuse_b=*/false);
            }
            const int nIdx = 16 * j + (lane & 15);
            #pragma unroll
            for (int v = 0; v < 8; ++v) O[(mbase + v) * NBINS + nIdx] = c[v];
        }
    }
    __syncthreads();

    if (lane < 16) {
        const int row = rt * 16 + lane;
        const float* W = &sh_w[lane][0];
        const float* S = &sh_s[lane][0];
        const float total = S[NBINS - 1];
        float best = -INFINITY;
        int bestT = 0;
        bool any = false;
        for (int n = 0; n < NBINS; ++n) {
            float w  = W[n], wf = 1.0f - w;
            float s  = S[n], sf = total - s;
            bool valid = (w > 0.0f) && (wf > 0.0f);
            float mb = s  / ((w  > 0.0f) ? w  : 1.0f);
            float mf = sf / ((wf > 0.0f) ? wf : 1.0f);
            float d  = mb - mf;
            float var = valid ? (w * wf * d * d) : -INFINITY;
            if (var > best) { best = var; bestT = n; }   // first max wins (strict >)
            any = any || valid;
        }
        const float mnv = ord2f(mmx[0]);
        const float mxv = ord2f(mmx[1]);
        int e = min(bestT + 1, NBINS - 1);
        float thr = mnv + (mxv - mnv) * ((float)e / 255.0f);   // edges = linspace(mn,mx,256)
        if (!any) thr = 0.0f;
        thrWs[row]  = thr;
        outThr[row] = thr;
    }
}

// ---------------- 4: apply threshold ----------------
__global__ void k_apply(const float4* __restrict__ x4, float4* __restrict__ o4,
                        const float* __restrict__ thr, int n4) {
    int stride = gridDim.x * blockDim.x;
    for (int i = blockIdx.x * blockDim.x + threadIdx.x; i < n4; i += stride) {
        int row = i >> 18;                 // HW/4 = 2^18 float4 per row
        float t = thr[row];
        float4 v = x4[i];
        v.x = (v.x <= t) ? 0.0f : v.x;
        v.y = (v.y <= t) ? 0.0f : v.y;
        v.z = (v.z <= t) ? 0.0f : v.z;
        v.w = (v.w <= t) ? 0.0f : v.w;
        o4[i] = v;
    }
}

extern "C" void kernel_launch(void* const* d_in, const int* in_sizes, int n_in,
                              void* d_out, int out_size, void* d_ws, size_t ws_size,
                              hipStream_t stream) {
    const float* x = (const float*)d_in[0];
    float* out = (float*)d_out;

    // workspace layout: [0..1] ordered min/max (pad to 8) | 48*256 hist | 48 thr
    unsigned* mmx   = (unsigned*)d_ws;
    unsigned* hist  = mmx + 8;
    float*    thrWs = (float*)(hist + NROW * NBINS);
    const int wsInts = 8 + NROW * NBINS + NROW;

    k_init<<<32, 256, 0, stream>>>(mmx, wsInts);
    k_minmax<<<4096, 256, 0, stream>>>((const float4*)x, mmx, (int)(NTOT / 4));
    k_hist<<<dim3(CHUNKS, NROW), 256, 0, stream>>>(x, mmx, hist);
    k_otsu<<<NROW / 16, 32, 0, stream>>>(hist, mmx, thrWs, out + NTOT);
    k_apply<<<8192, 256, 0, stream>>>((const float4*)x, (float4*)out, thrWs, (int)(NTOT / 4));
}